// LSTCWA_80745385165508
// MI455X (gfx1250) — compile-verified
//
#include <hip/hip_runtime.h>
#include <math.h>

#define BB 4
#define NN 8192
#define DD 512
#define LL 64
#define WW 64
#define SS 32
#define PP 128    // N / L
#define NWIN 4    // P / S

typedef __attribute__((ext_vector_type(2))) float v2f;
typedef __attribute__((ext_vector_type(8))) float v8f;

static __device__ __forceinline__ float warpRedAdd(float v) {
#pragma unroll
  for (int o = 16; o; o >>= 1) v += __shfl_xor(v, o, 32);
  return v;
}

// ---------------------------------------------------------------------------
// 1) Per-batch coord statistics: mean and std (ddof=1) over N, per component.
// ---------------------------------------------------------------------------
__global__ void k_cstats(const float* __restrict__ coords, float* __restrict__ cstat) {
  int b = blockIdx.x, t = threadIdx.x;
  float sx = 0, sy = 0, sxx = 0, syy = 0;
  for (int i = t; i < NN; i += 256) {
    float x = coords[(size_t)(b * NN + i) * 2 + 0];
    float y = coords[(size_t)(b * NN + i) * 2 + 1];
    sx += x; sy += y; sxx += x * x; syy += y * y;
  }
  __shared__ float r0[8], r1[8], r2[8], r3[8];
  sx = warpRedAdd(sx); sy = warpRedAdd(sy);
  sxx = warpRedAdd(sxx); syy = warpRedAdd(syy);
  int wave = t >> 5, lane = t & 31;
  if (lane == 0) { r0[wave] = sx; r1[wave] = sy; r2[wave] = sxx; r3[wave] = syy; }
  __syncthreads();
  if (t == 0) {
    float ax = 0, ay = 0, axx = 0, ayy = 0;
    for (int w = 0; w < 8; w++) { ax += r0[w]; ay += r1[w]; axx += r2[w]; ayy += r3[w]; }
    float mx = ax / NN, my = ay / NN;
    float vx = (axx - NN * mx * mx) / (NN - 1);
    float vy = (ayy - NN * my * my) / (NN - 1);
    cstat[b * 4 + 0] = mx;
    cstat[b * 4 + 1] = my;
    cstat[b * 4 + 2] = sqrtf(fmaxf(vx, 0.f)) + 1e-8f;
    cstat[b * 4 + 3] = sqrtf(fmaxf(vy, 0.f)) + 1e-8f;
  }
}

// ---------------------------------------------------------------------------
// 2) LayerNorm of feats (per token over D) + normalized coords. 1 wave/token.
// ---------------------------------------------------------------------------
__global__ void __launch_bounds__(256) k_ln(const float* __restrict__ feats,
                                            const float* __restrict__ coords,
                                            const float* __restrict__ cstat,
                                            float* __restrict__ fln,
                                            float* __restrict__ cn) {
  int token = blockIdx.x * 8 + (threadIdx.x >> 5);
  int lane = threadIdx.x & 31;
  const float* f = feats + (size_t)token * DD;
  float s = 0, ss = 0;
  for (int i = lane; i < DD; i += 32) { float v = f[i]; s += v; ss += v * v; }
  s = warpRedAdd(s); ss = warpRedAdd(ss);
  float mu = s / DD;
  float var = ss / DD - mu * mu;
  float r = rsqrtf(var + 1e-5f);
  float* o = fln + (size_t)token * DD;
  for (int i = lane; i < DD; i += 32) o[i] = (f[i] - mu) * r;
  if (lane == 0) {
    int b = token / NN;
    cn[(size_t)token * 2 + 0] = (coords[(size_t)token * 2 + 0] - cstat[b * 4 + 0]) / cstat[b * 4 + 2];
    cn[(size_t)token * 2 + 1] = (coords[(size_t)token * 2 + 1] - cstat[b * 4 + 1]) / cstat[b * 4 + 3];
  }
}

// ---------------------------------------------------------------------------
// 3) q = z @ Wq^T  (64 x 512, tiny)
// ---------------------------------------------------------------------------
__global__ void k_q(const float* __restrict__ z, const float* __restrict__ Wq,
                    float* __restrict__ q) {
  int l = blockIdx.x;
  __shared__ float zs[DD];
  for (int d = threadIdx.x; d < DD; d += 256) zs[d] = z[l * DD + d];
  __syncthreads();
  for (int e = threadIdx.x; e < DD; e += 256) {
    float a = 0;
    for (int d = 0; d < DD; d++) a += zs[d] * Wq[(size_t)e * DD + d];
    q[l * DD + e] = a;
  }
}

// ---------------------------------------------------------------------------
// 4) Contract q against Wk, W2, b2 once per l:
//    qk[l][d]=sum_e q[e]*Wk[e][d]; qw2[l][d]=sum_e q[e]*W2[e][d]; qb2[l]=q.b2
// ---------------------------------------------------------------------------
__global__ void k_qproj(const float* __restrict__ q, const float* __restrict__ Wk,
                        const float* __restrict__ W2, const float* __restrict__ b2,
                        float* __restrict__ qk, float* __restrict__ qw2,
                        float* __restrict__ qb2) {
  int l = blockIdx.x;
  __shared__ float qs[DD];
  for (int e = threadIdx.x; e < DD; e += 256) qs[e] = q[l * DD + e];
  __syncthreads();
  for (int d = threadIdx.x; d < DD; d += 256) {
    float a = 0, c = 0;
    for (int e = 0; e < DD; e++) {
      float qv = qs[e];
      a += qv * Wk[(size_t)e * DD + d];
      c += qv * W2[(size_t)e * DD + d];
    }
    qk[l * DD + d] = a;
    qw2[l * DD + d] = c;
  }
  float p = 0;
  for (int e = threadIdx.x; e < DD; e += 256) p += qs[e] * b2[e];
  p = warpRedAdd(p);
  __shared__ float rr[8];
  if ((threadIdx.x & 31) == 0) rr[threadIdx.x >> 5] = p;
  __syncthreads();
  if (threadIdx.x == 0) {
    float a = 0;
    for (int w = 0; w < 8; w++) a += rr[w];
    qb2[l] = a;
  }
}

// ---------------------------------------------------------------------------
// 5) Window kernel: one block per (b,l), loops the 4 windows.
//    logits = (f.qk + relu(h).qw2 + qb2)/sqrt(D); softmax; s += attn@f_win.
//    smean = mean over windows, plain store (deterministic).
// ---------------------------------------------------------------------------
__global__ void __launch_bounds__(256) k_window(const float* __restrict__ fln,
                                                const float* __restrict__ cn,
                                                const float* __restrict__ qk,
                                                const float* __restrict__ qw2,
                                                const float* __restrict__ qb2,
                                                const float* __restrict__ W1,
                                                const float* __restrict__ b1,
                                                float* __restrict__ smean) {
  int bx = blockIdx.x;               // 0..B*L-1
  int l = bx & (LL - 1), b = bx >> 6;
  int t = threadIdx.x;
  __shared__ float W1a[DD], W1b[DD], b1s[DD], qks[DD], qw2s[DD];
  __shared__ float px[WW], py[WW], logit[WW], attn[WW];
  __shared__ int gts[WW];
  __shared__ float mxy[2];

  for (int d = t; d < DD; d += 256) {
    W1a[d] = W1[d * 2 + 0];
    W1b[d] = W1[d * 2 + 1];
    b1s[d] = b1[d];
    qks[d] = qk[l * DD + d];
    qw2s[d] = qw2[l * DD + d];
  }
  float qb2v = qb2[l];
  float sacc0 = 0.f, sacc1 = 0.f;
  const int d0 = t, d1 = t + 256;

  for (int n = 0; n < NWIN; n++) {
    __syncthreads();  // protect shared reuse across window iterations
    if (t < WW) {
      int p = n * SS + t;
      if (p > PP - 1) p = PP - 1;  // idx_c clamp
      int g = ((b * LL) + l) * PP + p;
      gts[t] = g;
      px[t] = cn[(size_t)g * 2 + 0];
      py[t] = cn[(size_t)g * 2 + 1];
    }
    __syncthreads();
    int cnt = (PP - n * SS < WW) ? (PP - n * SS) : WW;  // valid window width
    if (t == 0) {
      float mx = 0, my = 0;
      for (int w = 0; w < cnt; w++) { mx += px[w]; my += py[w]; }
      mxy[0] = mx / cnt;
      mxy[1] = my / cnt;
    }
    __syncthreads();
    if (t < WW) { px[t] -= mxy[0]; py[t] -= mxy[1]; }
    __syncthreads();
    // logits: 4 threads per token, 128 channels each
    {
      int w = t >> 2, qd = t & 3;
      float pxx = px[w], pyy = py[w];
      const float* fr = fln + (size_t)gts[w] * DD;
      float acc = 0;
      int dlo = qd * 128;
      for (int d = dlo; d < dlo + 128; ++d) {
        float h = fmaxf(pxx * W1a[d] + pyy * W1b[d] + b1s[d], 0.f);
        acc += fr[d] * qks[d] + h * qw2s[d];
      }
      acc += __shfl_xor(acc, 1, 32);
      acc += __shfl_xor(acc, 2, 32);
      if (qd == 0) {
        float lg = (acc + qb2v) * 0.04419417382415922f;  // 1/sqrt(512)
        lg = fminf(fmaxf(lg, -10.f), 10.f);
        if (w >= cnt) lg = -3.0e38f;  // mask invalid -> -inf
        logit[w] = lg;
      }
    }
    __syncthreads();
    if (t < 32) {  // softmax over 64 by wave 0
      float a0 = logit[t], a1 = logit[t + 32];
      float m = fmaxf(a0, a1);
#pragma unroll
      for (int o = 16; o; o >>= 1) m = fmaxf(m, __shfl_xor(m, o, 32));
      float e0 = __expf(a0 - m), e1 = __expf(a1 - m);
      float s = warpRedAdd(e0 + e1);
      float inv = 1.f / s;
      attn[t] = e0 * inv;
      attn[t + 32] = e1 * inv;
    }
    __syncthreads();
    // s += attn @ f_win (two channels per thread)
    {
      float a0 = 0, a1 = 0;
      for (int w = 0; w < WW; w++) {
        float aw = attn[w];
        const float* fr = fln + (size_t)gts[w] * DD;
        a0 += aw * fr[d0];
        a1 += aw * fr[d1];
      }
      sacc0 += a0;
      sacc1 += a1;
    }
  }
  float* sm = smean + ((size_t)(b * LL) + l) * DD;
  sm[d0] = 0.25f * sacc0;  // mean over NWIN=4 windows
  sm[d1] = 0.25f * sacc1;
}

// ---------------------------------------------------------------------------
// 6) C[m][n] = sum_d A[m][d] * W[n][d] (+bias[n]) via V_WMMA_F32_16X16X4_F32.
//    Wave tile 16x64 (4 accumulators); block = 8 waves = 32x256 tile.
//    A lane layout: m = lane&15, k = k0 + 2*(lane>>4) + {0,1}
//    B lane layout: n = lane&15, same k pair (B[k][n] = W[n][k], contiguous).
//    C/D: VGPR r holds M = m0 + r + 8*(lane>>4), N = n0 + (lane&15).
// ---------------------------------------------------------------------------
__global__ void __launch_bounds__(256) k_gemm_wmma(const float* __restrict__ A,
                                                   const float* __restrict__ W,
                                                   const float* __restrict__ bias,
                                                   float* __restrict__ C) {
  int t = threadIdx.x, lane = t & 31, wave = t >> 5;
  int m0 = blockIdx.x * 32 + (wave & 1) * 16;
  int n0 = blockIdx.y * 256 + (wave >> 1) * 64;
  int ml = lane & 15;
  int kh = (lane >> 4) * 2;
  const float* Arow = A + (size_t)(m0 + ml) * DD;
  const float* Wr0 = W + (size_t)(n0 + 0 + ml) * DD;
  const float* Wr1 = W + (size_t)(n0 + 16 + ml) * DD;
  const float* Wr2 = W + (size_t)(n0 + 32 + ml) * DD;
  const float* Wr3 = W + (size_t)(n0 + 48 + ml) * DD;
  v8f c0 = {}, c1 = {}, c2 = {}, c3 = {};
  for (int k0 = 0; k0 < DD; k0 += 4) {
    v2f a = *(const v2f*)(Arow + k0 + kh);
    v2f b0 = *(const v2f*)(Wr0 + k0 + kh);
    v2f b1v = *(const v2f*)(Wr1 + k0 + kh);
    v2f b2v = *(const v2f*)(Wr2 + k0 + kh);
    v2f b3v = *(const v2f*)(Wr3 + k0 + kh);
    c0 = __builtin_amdgcn_wmma_f32_16x16x4_f32(false, a, false, b0, (short)0, c0, false, false);
    c1 = __builtin_amdgcn_wmma_f32_16x16x4_f32(false, a, false, b1v, (short)0, c1, false, false);
    c2 = __builtin_amdgcn_wmma_f32_16x16x4_f32(false, a, false, b2v, (short)0, c2, false, false);
    c3 = __builtin_amdgcn_wmma_f32_16x16x4_f32(false, a, false, b3v, (short)0, c3, false, false);
  }
  float bi0 = bias ? bias[n0 + 0 + ml] : 0.f;
  float bi1 = bias ? bias[n0 + 16 + ml] : 0.f;
  float bi2 = bias ? bias[n0 + 32 + ml] : 0.f;
  float bi3 = bias ? bias[n0 + 48 + ml] : 0.f;
  int mbase = m0 + ((lane >> 4) << 3);
#pragma unroll
  for (int r = 0; r < 8; r++) {
    size_t row = (size_t)(mbase + r) * DD;
    C[row + n0 + 0 + ml] = c0[r] + bi0;
    C[row + n0 + 16 + ml] = c1[r] + bi1;
    C[row + n0 + 32 + ml] = c2[r] + bi2;
    C[row + n0 + 48 + ml] = c3[r] + bi3;
  }
}

// ---------------------------------------------------------------------------
extern "C" void kernel_launch(void* const* d_in, const int* in_sizes, int n_in,
                              void* d_out, int out_size, void* d_ws, size_t ws_size,
                              hipStream_t stream) {
  (void)in_sizes; (void)n_in; (void)out_size; (void)ws_size;
  const float* feats  = (const float*)d_in[0];
  const float* coords = (const float*)d_in[1];
  // d_in[2] = mask: unused by the reference computation (all false)
  const float* z  = (const float*)d_in[3];
  const float* Wq = (const float*)d_in[4];
  const float* Wk = (const float*)d_in[5];
  const float* Wv = (const float*)d_in[6];
  const float* W1 = (const float*)d_in[7];
  const float* b1 = (const float*)d_in[8];
  const float* W2 = (const float*)d_in[9];
  const float* b2 = (const float*)d_in[10];
  const float* Wo = (const float*)d_in[11];
  const float* bo = (const float*)d_in[12];
  float* out = (float*)d_out;

  char* ws = (char*)d_ws;
  size_t off = 0;
  float* fln   = (float*)(ws + off); off += (size_t)BB * NN * DD * 4;  // 64 MB
  float* cnb   = (float*)(ws + off); off += (size_t)BB * NN * 2 * 4;   // 256 KB
  float* cstat = (float*)(ws + off); off += 256;
  float* q     = (float*)(ws + off); off += (size_t)LL * DD * 4;
  float* qk    = (float*)(ws + off); off += (size_t)LL * DD * 4;
  float* qw2   = (float*)(ws + off); off += (size_t)LL * DD * 4;
  float* qb2   = (float*)(ws + off); off += 256;
  float* smean = (float*)(ws + off); off += (size_t)BB * LL * DD * 4;  // 512 KB
  float* tmp   = (float*)(ws + off); off += (size_t)BB * LL * DD * 4;  // 512 KB

  k_cstats<<<BB, 256, 0, stream>>>(coords, cstat);
  k_ln<<<(BB * NN) / 8, 256, 0, stream>>>(feats, coords, cstat, fln, cnb);
  k_q<<<LL, 256, 0, stream>>>(z, Wq, q);
  k_qproj<<<LL, 256, 0, stream>>>(q, Wk, W2, b2, qk, qw2, qb2);
  k_window<<<BB * LL, 256, 0, stream>>>(fln, cnb, qk, qw2, qb2, W1, b1, smean);
  // out = ((smean @ Wv^T) @ Wo^T) + bo   -- f32 WMMA, full precision
  k_gemm_wmma<<<dim3((BB * LL) / 32, DD / 256), 256, 0, stream>>>(smean, Wv, nullptr, tmp);
  k_gemm_wmma<<<dim3((BB * LL) / 32, DD / 256), 256, 0, stream>>>(tmp, Wo, bo, out);
}